// ChannelMaxPooling_38534446579961
// MI455X (gfx1250) — compile-verified
//
#include <hip/hip_runtime.h>
#include <stdint.h>

// Problem constants (fixed by the reference: inputs [1024,7,7,512] f32, OUT_PLANES=512)
#define S7        7
#define CCH       512
#define PIX       (S7 * S7)          // 49
#define BATCH     1024
#define TASKS     (BATCH * PIX)      // 50176
#define K_PP      10                 // 512 / 49
#define K_C       22                 // 512 % 49
#define OUTP      512
#define CENTERPIX 24                 // 3*7 + 3

#define WPB       8                  // waves per block (256 threads, wave32)
#define BLOCKSZ   256
#define NWAVES    (TASKS / 2)        // 25088 waves, exactly 2 pixels each
#define NBLOCKS   (NWAVES / WPB)     // 3136

typedef uint32_t u32;
typedef __attribute__((ext_vector_type(4))) u32   v4u;
typedef __attribute__((ext_vector_type(8))) int   v8i;
typedef __attribute__((ext_vector_type(4))) int   v4i;
typedef __attribute__((ext_vector_type(4))) float v4f;

// ---------------------------------------------------------------------------
// TDM: DMA one contiguous 2KB pixel row (512 f32) from global into LDS.
// Descriptor per CDNA5 ISA (D# groups 0/1; groups 2/3
// zero since the tensor is <=2D). Tracked by TENSORcnt.
// This toolchain exposes the 6-arg builtin:
//   (uint32x4 g0, int32x8 g1, int32x4, int32x4, int32x8, i32 cpol)
// ---------------------------------------------------------------------------
__device__ __forceinline__ void tdm_load_row(const float* gsrc, u32 lds_off) {
    u32 glo = __builtin_amdgcn_readfirstlane((u32)(uintptr_t)gsrc);
    u32 ghi = __builtin_amdgcn_readfirstlane((u32)((uintptr_t)gsrc >> 32));
    lds_off = __builtin_amdgcn_readfirstlane(lds_off);

    v4u g0;
    g0.x = 1u;                                   // [1:0] count=1 (valid), user mode
    g0.y = lds_off;                              // [63:32]  lds_addr (bytes)
    g0.z = glo;                                  // [95:64]  global_addr lo
    g0.w = (ghi & 0x01FFFFFFu) | (2u << 30);     // [120:96] global_addr hi, [127:126] type=2

    v8i g1;
    g1[0] = (2 << 16);            // [17:16] data_size=2 -> 4 bytes; wg_mask=0
    g1[1] = (int)(512u << 16);    // [79:48] tensor_dim0 = 512 (low 16 bits here)
    g1[2] = (1 << 16);            // [111:80] tensor_dim1 = 1
    g1[3] = (int)(512u << 16);    // [127:112] tile_dim0 = 512
    g1[4] = 1;                    // [143:128] tile_dim1 = 1; tile_dim2 = 0
    g1[5] = 512;                  // [207:160] tensor_dim0_stride = 512
    g1[6] = 0;
    g1[7] = 0;

    v4i z4 = {0, 0, 0, 0};
    v8i z8 = {0, 0, 0, 0, 0, 0, 0, 0};
    __builtin_amdgcn_tensor_load_to_lds(g0, g1, z4, z4, z8, 0);
}

// ---------------------------------------------------------------------------
// One wave computes sorted top-k of one pixel's 512 channels (from LDS).
// Per-lane: 16 values, fully sorted descending by a Batcher odd-even
// mergesort (63 CEs, unmasked max/min pairs -> VOPD-friendly). Then k rounds
// of 5-step shfl_xor arg-max (lane-id tie-break => deterministic under
// duplicates); the winning lane pops its sorted list (register shift).
// ---------------------------------------------------------------------------
__device__ __forceinline__ void topk_pixel(const float* __restrict__ buf,
                                           int task, float* __restrict__ out,
                                           int lane) {
    float v[16];
#pragma unroll
    for (int c = 0; c < 4; ++c) {
        v4f q = *(const v4f*)(buf + (c * 32 + lane) * 4);   // ds_load_b128
        v[c * 4 + 0] = q.x; v[c * 4 + 1] = q.y;
        v[c * 4 + 2] = q.z; v[c * 4 + 3] = q.w;
    }

    // Batcher odd-even mergesort, n=16, descending (v[0] = lane max).
#pragma unroll
    for (int p = 1; p < 16; p <<= 1) {
#pragma unroll
        for (int k2 = p; k2 >= 1; k2 >>= 1) {
#pragma unroll
            for (int j = k2 % p; j + k2 < 16; j += 2 * k2) {
#pragma unroll
                for (int i = 0; i < k2; ++i) {
                    if (i + j + k2 < 16) {
                        if ((i + j) / (2 * p) == (i + j + k2) / (2 * p)) {
                            const int a = i + j, b = i + j + k2;
                            const float hi = fmaxf(v[a], v[b]);
                            const float lo = fminf(v[a], v[b]);
                            v[a] = hi; v[b] = lo;
                        }
                    }
                }
            }
        }
    }

    const int b   = task / PIX;
    const int pix = task - b * PIX;
    const int k   = (pix == CENTERPIX) ? K_C : K_PP;

    float res = 0.0f;
    for (int it = 0; it < k; ++it) {
        float w = v[0];
        int   wl = lane;
#pragma unroll
        for (int off = 16; off > 0; off >>= 1) {
            const float ow  = __shfl_xor(w, off, 32);
            const int   owl = __shfl_xor(wl, off, 32);
            const bool take = (ow > w) || ((ow == w) && (owl < wl));
            w  = take ? ow : w;
            wl = take ? owl : wl;
        }
        if (lane == it) res = w;                 // lane it keeps the it-th largest
        if (lane == wl) {                        // winner pops its sorted list
#pragma unroll
            for (int i = 0; i < 15; ++i) v[i] = v[i + 1];
            v[15] = -__builtin_huge_valf();
        }
    }

    const size_t ob = (size_t)b * OUTP;
    if (pix == CENTERPIX) {
        if (lane < K_C)  out[ob + lane] = res;                              // center top-22
        if (lane < K_PP) out[ob + K_C + CENTERPIX * K_PP + lane] = res;     // its per-pixel top-10
    } else {
        if (lane < K_PP) out[ob + K_C + pix * K_PP + lane] = res;
    }
}

// ---------------------------------------------------------------------------
// 25088 waves x 2 pixels, double-buffered TDM pipeline:
//   issue DMA(t0), issue DMA(t1), wait cnt<=1, compute t0, wait 0, compute t1
// LDS: 8 waves * 2 buffers * 2KB = 32KB per block (<< 320KB/WGP).
// ---------------------------------------------------------------------------
__global__ void __launch_bounds__(BLOCKSZ)
topk_tdm_kernel(const float* __restrict__ in, float* __restrict__ out) {
    __shared__ float lds_buf[WPB * 2 * CCH];

    const int tid  = threadIdx.x;
    const int wid  = tid >> 5;
    const int lane = tid & 31;
    const int g    = blockIdx.x * WPB + wid;     // wave id: 0 .. NWAVES-1

    float* buf0 = &lds_buf[(wid * 2 + 0) * CCH];
    float* buf1 = &lds_buf[(wid * 2 + 1) * CCH];

    const int t0 = g;
    const int t1 = g + NWAVES;

    tdm_load_row(in + (size_t)t0 * CCH, (u32)(uintptr_t)buf0);
    tdm_load_row(in + (size_t)t1 * CCH, (u32)(uintptr_t)buf1);

    __builtin_amdgcn_s_wait_tensorcnt(1);        // first DMA complete (in-order)
    topk_pixel(buf0, t0, out, lane);

    __builtin_amdgcn_s_wait_tensorcnt(0);        // second DMA complete
    topk_pixel(buf1, t1, out, lane);
}

extern "C" void kernel_launch(void* const* d_in, const int* in_sizes, int n_in,
                              void* d_out, int out_size, void* d_ws, size_t ws_size,
                              hipStream_t stream) {
    (void)in_sizes; (void)n_in; (void)out_size; (void)d_ws; (void)ws_size;
    const float* in  = (const float*)d_in[0];
    float*       out = (float*)d_out;
    topk_tdm_kernel<<<NBLOCKS, BLOCKSZ, 0, stream>>>(in, out);
}